// SSIMLoss_17918603558951
// MI455X (gfx1250) — compile-verified
//
#include <hip/hip_runtime.h>

typedef float v2f __attribute__((ext_vector_type(2)));
typedef float v8f __attribute__((ext_vector_type(8)));

#define IMG_H 512
#define IMG_W 512
#define NIMG  32
#define SSIM_C1 1.0e-4f
#define SSIM_C2 9.0e-4f
#define WAVES_PER_BLOCK 8
#define NTILES (NIMG * 32 * 32)            // 32768
#define NBLOCKS (NTILES / WAVES_PER_BLOCK) // 4096
#define NPIX 8388608.0f                    // 32*512*512

__device__ __forceinline__ v8f wmma4(v2f a, v2f b, v8f c) {
  // D = A(16x4) * B(4x16) + C, f32.
  // (neg_a, A, neg_b, B, c_mod, C, reuse_a, reuse_b)
  return __builtin_amdgcn_wmma_f32_16x16x4_f32(false, a, false, b, (short)0, c,
                                               false, false);
}

// lane <-> lane^16 swap as pure-VALU v_permlanex16 (identity lane select).
__device__ __forceinline__ float swap16(float v) {
  int i = __builtin_bit_cast(int, v);
  int r = __builtin_amdgcn_permlanex16(i, i, 0x76543210, 0xfedcba98, false, false);
  return __builtin_bit_cast(float, r);
}

// 2D 5x5 box filter (1/25 via 1/5 per pass) of a 20x20 padded tile.
// A1 = padded rows 0..15 in A-layout chunks, A2 = padded rows 16..19 (rest zero).
__device__ __forceinline__ v8f box5(const v2f* A1, const v2f* A2,
                                    const v2f* band, bool lo) {
  v8f h1 = {0.f, 0.f, 0.f, 0.f, 0.f, 0.f, 0.f, 0.f};
  v8f h2 = h1;
#pragma unroll
  for (int k = 0; k < 5; ++k) {       // H = P * Bh   (K = 20 in chunks of 4)
    h1 = wmma4(A1[k], band[k], h1);
    h2 = wmma4(A2[k], band[k], h2);
  }
  // Rearrange H (D-layout: VGPR v = rows v / v+8 in lane halves) into B-layout
  // chunks (VGPR pair = rows q,q+1 lo-half / q+2,q+3 hi-half).
  float t0 = swap16(h1[0]), t1 = swap16(h1[1]), t2 = swap16(h1[2]), t3 = swap16(h1[3]);
  float t4 = swap16(h1[4]), t5 = swap16(h1[5]), t6 = swap16(h1[6]), t7 = swap16(h1[7]);
  float u2 = swap16(h2[2]), u3 = swap16(h2[3]);

  v8f out = {0.f, 0.f, 0.f, 0.f, 0.f, 0.f, 0.f, 0.f};
  v2f b0 = {lo ? h1[0] : t2, lo ? h1[1] : t3};   // rows 0..3
  out = wmma4(band[0], b0, out);
  v2f b1 = {lo ? h1[4] : t6, lo ? h1[5] : t7};   // rows 4..7
  out = wmma4(band[1], b1, out);
  v2f b2 = {lo ? t0 : h1[2], lo ? t1 : h1[3]};   // rows 8..11
  out = wmma4(band[2], b2, out);
  v2f b3 = {lo ? t4 : h1[6], lo ? t5 : h1[7]};   // rows 12..15
  out = wmma4(band[3], b3, out);
  v2f b4 = {lo ? h2[0] : u2, lo ? h2[1] : u3};   // rows 16..19
  out = wmma4(band[4], b4, out);                 // Out = Bv * H
  return out;
}

extern "C" __global__ __launch_bounds__(256) void ssim_map_kernel(
    const float* __restrict__ img1, const float* __restrict__ img2,
    float* __restrict__ partials) {
  const int lane = threadIdx.x & 31;
  const int wave = threadIdx.x >> 5;
  const int tile = blockIdx.x * WAVES_PER_BLOCK + wave;
  const int img = tile >> 10;         // 32x32 tiles per image
  const int ty = (tile >> 5) & 31;
  const int tx = tile & 31;
  const size_t ibase = (size_t)img * (IMG_H * IMG_W);
  const float* p1 = img1 + ibase;
  const float* p2 = img2 + ibase;
  const int R0 = ty << 4, C0 = tx << 4;
  const int m = lane & 15;
  const bool lo = lane < 16;
  const int koff = lo ? 0 : 2;

  // Banded 1/5 matrix: val(row i, k j) = 1/5 if 0 <= j - i <= 4.
  // Pinned in VGPRs so the cmp/cndmask chains aren't rematerialized.
  v2f band[5];
#pragma unroll
  for (int k = 0; k < 5; ++k) {
    int j = 4 * k + koff;
    int d0 = j - m, d1 = j + 1 - m;
    float bx = (d0 >= 0 && d0 <= 4) ? 0.2f : 0.0f;
    float by = (d1 >= 0 && d1 <= 4) ? 0.2f : 0.0f;
    asm("" : "+v"(bx));
    asm("" : "+v"(by));
    band[k].x = bx;
    band[k].y = by;
  }

  // Load padded 20x20 tiles of img1/img2 directly in A-layout.
  // Lane l (<16): row l, cols {4k, 4k+1}; lane l+16: row l, cols {4k+2, 4k+3}.
  // Zero padding via clamped in-image addresses + 0/1 float masks (branch-free):
  // pairs are even-aligned so the pad-2 border never straddles a pair.
  const int r1 = R0 - 2 + m;               // padded rows 0..15
  const int r1c = min(max(r1, 0), IMG_H - 1);
  const float mR1 = ((unsigned)r1 < (unsigned)IMG_H) ? 1.f : 0.f;
  const int r2 = R0 + 14 + m;              // padded rows 16..19 (m<4)
  const int r2c = min(max(r2, 0), IMG_H - 1);
  const float mR2 = (((unsigned)r2 < (unsigned)IMG_H) && (m < 4)) ? 1.f : 0.f;
  const int rb1 = r1c * IMG_W;
  const int rb2 = r2c * IMG_W;
  const int gbase = C0 - 2 + koff;

  v2f x1[5], y1[5], x2[5], y2[5];
#pragma unroll
  for (int k = 0; k < 5; ++k) {
    const int g = gbase + 4 * k;           // even global column
    const int gc = min(max(g, 0), IMG_W - 2);
    const float mG = ((unsigned)g < (unsigned)IMG_W) ? 1.f : 0.f;
    const float mk1 = mR1 * mG;
    const float mk2 = mR2 * mG;
    const int o1 = rb1 + gc;
    const int o2 = rb2 + gc;
    x1[k] = *(const v2f*)(p1 + o1) * mk1;
    y1[k] = *(const v2f*)(p2 + o1) * mk1;
    x2[k] = *(const v2f*)(p1 + o2) * mk2;
    y2[k] = *(const v2f*)(p2 + o2) * mk2;
  }

  // Five box-filtered channels; products are elementwise in A-layout.
  v8f mu1 = box5(x1, x2, band, lo);
  v8f mu2 = box5(y1, y2, band, lo);
  v2f q1[5], q2[5];
#pragma unroll
  for (int k = 0; k < 5; ++k) { q1[k] = x1[k] * x1[k]; q2[k] = x2[k] * x2[k]; }
  v8f e11 = box5(q1, q2, band, lo);
#pragma unroll
  for (int k = 0; k < 5; ++k) { q1[k] = y1[k] * y1[k]; q2[k] = y2[k] * y2[k]; }
  v8f e22 = box5(q1, q2, band, lo);
#pragma unroll
  for (int k = 0; k < 5; ++k) { q1[k] = x1[k] * y1[k]; q2[k] = x2[k] * y2[k]; }
  v8f e12 = box5(q1, q2, band, lo);

  // Pointwise SSIM + accumulate (each of the 256 tile pixels appears once).
  float acc = 0.f;
#pragma unroll
  for (int e = 0; e < 8; ++e) {
    float m1 = mu1[e], m2 = mu2[e];
    float s1 = e11[e] - m1 * m1;
    float s2 = e22[e] - m2 * m2;
    float s12 = e12[e] - m1 * m2;
    float num = (2.f * m1 * m2 + SSIM_C1) * (2.f * s12 + SSIM_C2);
    float den = (m1 * m1 + m2 * m2 + SSIM_C1) * (s1 + s2 + SSIM_C2);
    acc += num * __builtin_amdgcn_rcpf(den);
  }
  // Wave reduction: lane^16 via permlanex16 (VALU), rest via shfl.
  acc += swap16(acc);
#pragma unroll
  for (int off = 8; off >= 1; off >>= 1) acc += __shfl_xor(acc, off, 32);

  __shared__ float sred[WAVES_PER_BLOCK];
  if (lane == 0) sred[wave] = acc;
  __syncthreads();
  if (threadIdx.x == 0) {
    float s = 0.f;
#pragma unroll
    for (int i = 0; i < WAVES_PER_BLOCK; ++i) s += sred[i];
    partials[blockIdx.x] = s;
  }
}

extern "C" __global__ __launch_bounds__(256) void ssim_finalize_kernel(
    const float* __restrict__ partials, int n, float* __restrict__ out) {
  __shared__ float sm[256];
  float s = 0.f;
  for (int i = threadIdx.x; i < n; i += 256) s += partials[i];
  sm[threadIdx.x] = s;
  __syncthreads();
  for (int st = 128; st >= 1; st >>= 1) {
    if ((int)threadIdx.x < st) sm[threadIdx.x] += sm[threadIdx.x + st];
    __syncthreads();
  }
  if (threadIdx.x == 0) out[0] = 1.0f - sm[0] / NPIX;
}

extern "C" void kernel_launch(void* const* d_in, const int* in_sizes, int n_in,
                              void* d_out, int out_size, void* d_ws, size_t ws_size,
                              hipStream_t stream) {
  (void)in_sizes; (void)n_in; (void)out_size; (void)ws_size;
  const float* img1 = (const float*)d_in[0];
  const float* img2 = (const float*)d_in[1];
  float* partials = (float*)d_ws;           // NBLOCKS floats of scratch
  float* out = (float*)d_out;

  ssim_map_kernel<<<NBLOCKS, 256, 0, stream>>>(img1, img2, partials);
  ssim_finalize_kernel<<<1, 256, 0, stream>>>(partials, NBLOCKS, out);
}